// MiddleBlock_71820443124035
// MI455X (gfx1250) — compile-verified
//
#include <hip/hip_runtime.h>
#include <hip/hip_bf16.h>

typedef __attribute__((ext_vector_type(2))) float v2f;
typedef __attribute__((ext_vector_type(8))) float v8f;

#define NB_   8
#define FIN_  64
#define FOUT_ 128
#define NN_   196608
#define BB_   2

// out[node,f] = scale * ( diag[n]*Y[node,f] + sum_j ew[n,j]*Y[b, nbr[n,j], f] ) - Prev[node,f]
// One thread per (node, feature). Neighbor rows are contiguous (ldY floats apart).
__global__ __launch_bounds__(256) void cheb_gather(
    const float* __restrict__ Y, unsigned ldY,
    const float* __restrict__ Prev, unsigned ldPrev,
    const int* __restrict__ nbr, const float* __restrict__ ew,
    const float* __restrict__ diag, float scale,
    float* __restrict__ Out, float* __restrict__ SelfOut)
{
    size_t gid  = (size_t)blockIdx.x * 256 + threadIdx.x;
    int    f    = (int)(gid & (FIN_ - 1));
    size_t node = gid >> 6;                       // b*N + n
    size_t n    = (node >= (size_t)NN_) ? node - (size_t)NN_ : node;
    size_t bb   = node - n;                       // b*N

    float self = Y[node * (size_t)ldY + f];
    float acc  = diag[n] * self;
#pragma unroll
    for (int j = 0; j < NB_; ++j) {
        int   nb = nbr[n * NB_ + j];
        float w  = ew[n * NB_ + j];
        acc = fmaf(w, Y[(bb + (size_t)nb) * (size_t)ldY + f], acc);
    }
    float r = scale * acc;
    if (Prev) r -= Prev[node * (size_t)ldPrev + f];
    Out[node * 256 + f] = r;                      // H row stride = 256 floats
    if (SelfOut) SelfOut[node * 256 + f] = self;  // stash x into H[:,0:64]
}

// C(16 x 128) = A(16 x KTOT) * W(KTOT x 128); then +bias, relu, LayerNorm(128) -> Out.
// 256 threads = 8 wave32; wave w owns output columns [16w, 16w+16).
template <int KTOT>
__global__ __launch_bounds__(256) void gemm_bias_relu_ln(
    const float* __restrict__ A, const float* __restrict__ W,
    const float* __restrict__ bias, const float* __restrict__ gamma,
    const float* __restrict__ beta, float* __restrict__ Out)
{
    constexpr int LDA = KTOT + 4;   // pad: conflict-free ds_load_b64 A fragments
    constexpr int LDC = FOUT_ + 4;  // pad: conflict-free C staging
    __shared__ float smem[16 * LDA];

    const size_t row0 = (size_t)blockIdx.x * 16;

    // Cooperative A-tile load (float4, coalesced).
    constexpr int K4 = KTOT / 4;
    for (int t = threadIdx.x; t < 16 * K4; t += 256) {
        int r = t / K4, c4 = t - r * K4;
        const float4 v = ((const float4*)(A + (row0 + (size_t)r) * KTOT))[c4];
        *(float4*)&smem[r * LDA + c4 * 4] = v;
    }
    __syncthreads();

    const int lane = threadIdx.x & 31;
    const int wave = threadIdx.x >> 5;
    const int n0   = wave * 16;
    const int mrow = lane & 15;
    const int koff = (lane >> 4) << 1;   // lanes 16-31 hold K+2/K+3

    const float* arow = &smem[mrow * LDA + koff];
    const float* wcol = W + (size_t)koff * FOUT_ + n0 + mrow;

    v8f c = {};
#pragma unroll 8
    for (int k = 0; k < KTOT; k += 4) {
        v2f a = *(const v2f*)(arow + k);                 // A[m][k0..k0+1]
        v2f b;
        b.x = wcol[(size_t)k * FOUT_];                   // W[k0  ][n]
        b.y = wcol[(size_t)k * FOUT_ + FOUT_];           // W[k0+1][n]
        c = __builtin_amdgcn_wmma_f32_16x16x4_f32(
                false, a, false, b, (short)0, c, false, false);
    }

    __syncthreads();   // all waves done reading A-tile; reuse smem for C
    {
        float bc = bias[n0 + mrow];
#pragma unroll
        for (int r = 0; r < 8; ++r) {
            int   m = r + ((lane >> 4) << 3);            // C/D layout: m = r + 8*(lane>=16)
            float v = fmaxf(c[r] + bc, 0.0f);            // bias + relu
            smem[m * LDC + n0 + mrow] = v;
        }
    }
    __syncthreads();

    // LayerNorm over 128 columns: wave handles rows 2*wave, 2*wave+1.
#pragma unroll
    for (int rr = 0; rr < 2; ++rr) {
        int   m  = wave * 2 + rr;
        float v0 = smem[m * LDC + lane +  0];
        float v1 = smem[m * LDC + lane + 32];
        float v2 = smem[m * LDC + lane + 64];
        float v3 = smem[m * LDC + lane + 96];
        float s  = v0 + v1 + v2 + v3;
        float q  = v0 * v0 + v1 * v1 + v2 * v2 + v3 * v3;
#pragma unroll
        for (int off = 16; off > 0; off >>= 1) {
            s += __shfl_xor(s, off, 32);
            q += __shfl_xor(q, off, 32);
        }
        float mean = s * (1.0f / FOUT_);
        float var  = q * (1.0f / FOUT_) - mean * mean;
        float rs   = rsqrtf(var + 1e-6f);
        size_t orow = (row0 + (size_t)m) * FOUT_;
        float  vv[4] = {v0, v1, v2, v3};
#pragma unroll
        for (int i = 0; i < 4; ++i) {
            int col = lane + 32 * i;
            Out[orow + col] = (vv[i] - mean) * rs * gamma[col] + beta[col];
        }
    }
}

extern "C" void kernel_launch(void* const* d_in, const int* in_sizes, int n_in,
                              void* d_out, int out_size, void* d_ws, size_t ws_size,
                              hipStream_t stream) {
    const float* x      = (const float*)d_in[0];
    const int*   nbr    = (const int*)  d_in[1];
    const float* ew     = (const float*)d_in[2];
    const float* diag   = (const float*)d_in[3];
    const float* cheb_w = (const float*)d_in[4];
    const float* cheb_b = (const float*)d_in[5];
    const float* g1     = (const float*)d_in[6];
    const float* b1     = (const float*)d_in[7];
    const float* pc_w   = (const float*)d_in[8];
    const float* pc_b   = (const float*)d_in[9];
    const float* g2     = (const float*)d_in[10];
    const float* b2     = (const float*)d_in[11];

    const size_t M1 = (size_t)BB_ * NN_;          // 393216 rows
    float* H    = (float*)d_ws;                   // M1 x 256  (x | T1 | T2 | T3)
    float* Hmid = H + M1 * 256;                   // M1 x 128

    dim3 blk(256);
    dim3 gG((unsigned)(M1 * FIN_ / 256));         // 98304 blocks: one thread per (node,f)

    // T1 = L(x); also copy x into H[:,0:64]
    hipLaunchKernelGGL(cheb_gather, gG, blk, 0, stream,
        x, (unsigned)FIN_, (const float*)nullptr, 0u,
        nbr, ew, diag, 1.0f, H + 64, H);
    // T2 = 2*L(T1) - x
    hipLaunchKernelGGL(cheb_gather, gG, blk, 0, stream,
        H + 64, 256u, x, (unsigned)FIN_,
        nbr, ew, diag, 2.0f, H + 128, (float*)nullptr);
    // T3 = 2*L(T2) - T1
    hipLaunchKernelGGL(cheb_gather, gG, blk, 0, stream,
        H + 128, 256u, H + 64, 256u,
        nbr, ew, diag, 2.0f, H + 192, (float*)nullptr);

    // GEMM1 + bias + relu + LN  -> Hmid (viewed later as M1/4 x 512)
    hipLaunchKernelGGL((gemm_bias_relu_ln<256>), dim3((unsigned)(M1 / 16)), blk, 0, stream,
        H, cheb_w, cheb_b, g1, b1, Hmid);
    // GEMM2 + bias + relu + LN  -> d_out
    hipLaunchKernelGGL((gemm_bias_relu_ln<512>), dim3((unsigned)(M1 / 4 / 16)), blk, 0, stream,
        Hmid, pc_w, pc_b, g2, b2, (float*)d_out);
}